// TGCNCell_19628000542754
// MI455X (gfx1250) — compile-verified
//
#include <hip/hip_runtime.h>
#include <math.h>
#include <stdint.h>

typedef __attribute__((ext_vector_type(2))) float v2f;
typedef __attribute__((ext_vector_type(8))) float v8f;
typedef __attribute__((ext_vector_type(4))) int   v4i;

#define FDIM 128   // hidden / feature width of every GEMM output
#define KC   64    // K-chunk staged in LDS (64 x 128 fp32 = 32 KB)

// ---------------------------------------------------------------- degree prep
__global__ void init_deg_kernel(float* __restrict__ deg, int n) {
  int i = blockIdx.x * blockDim.x + threadIdx.x;
  if (i < n) deg[i] = 1.0f;  // self-loop weight
}

__global__ void scatter_deg_kernel(const long long* __restrict__ ei,
                                   const float* __restrict__ w,
                                   float* __restrict__ deg, int E) {
  int e = blockIdx.x * blockDim.x + threadIdx.x;
  if (e < E) {
    int d = (int)ei[(size_t)E + e];   // dst row of edge_index
    unsafeAtomicAdd(&deg[d], w[e]);
  }
}

__global__ void rsqrt_kernel(const float* __restrict__ deg,
                             float* __restrict__ dinv, int n) {
  int i = blockIdx.x * blockDim.x + threadIdx.x;
  if (i < n) dinv[i] = rsqrtf(deg[i]);
}

__global__ void zero4_kernel(float4* __restrict__ p, long n4) {
  long i = (long)blockIdx.x * blockDim.x + threadIdx.x;
  if (i < n4) p[i] = make_float4(0.f, 0.f, 0.f, 0.f);
}

// ---------------------------------------------------------------- fp32 WMMA GEMM
// C[M x 128] = A[M x K] * W[K x 128], row-major, M % 16 == 0.
// 256-thread block = 8 waves; each wave computes a 16 x 128 strip with
// v_wmma_f32_16x16x4_f32. Weights staged through LDS in 64-row chunks
// (async global->LDS, ASYNCcnt-tracked), shared by all 8 waves.
__global__ void __launch_bounds__(256)
gemm_wmma_f32_kernel(const float* __restrict__ A, const float* __restrict__ W,
                     float* __restrict__ C, int M, int K) {
  __shared__ float lw[KC * FDIM];          // 32 KB weight chunk

  const int wave = threadIdx.x >> 5;
  int tile = blockIdx.x * 8 + wave;
  const bool valid = (tile * 16 < M);
  if (!valid) tile = 0;                    // clamp: keep barriers/EXEC uniform
  const int lane = threadIdx.x & 31;
  const int l    = lane & 15;              // row (A) / col (B,C) within tile
  const int half = lane >> 4;              // selects K pair / M upper half
  const long r0  = (long)tile * 16;

  v8f acc[8];
#pragma unroll
  for (int nt = 0; nt < 8; ++nt) acc[nt] = (v8f)(0.0f);

  const float* arow = A + (r0 + l) * (long)K + 2 * half;

  for (int kc = 0; kc < K; kc += KC) {
    __syncthreads();                       // previous chunk fully consumed
    // ---- stage W[kc : kc+KC][0:128] into LDS (8192 floats, 16B per lane op)
#if __has_builtin(__builtin_amdgcn_global_load_async_to_lds_b128) && \
    __has_builtin(__builtin_amdgcn_s_wait_asynccnt)
    {
      __attribute__((address_space(1))) v4i* gsrc =
          (__attribute__((address_space(1))) v4i*)(void*)(
              const_cast<float*>(W) + (long)kc * FDIM);
      __attribute__((address_space(3))) v4i* ldst =
          (__attribute__((address_space(3))) v4i*)(void*)lw;
#pragma unroll
      for (int t = 0; t < 8; ++t) {
        int idx = threadIdx.x + t * 256;   // v4i element index (16B units)
        __builtin_amdgcn_global_load_async_to_lds_b128(gsrc + idx, ldst + idx, 0, 0);
      }
      __builtin_amdgcn_s_wait_asynccnt(0);
    }
#else
    {
      const float4* src = (const float4*)(W + (long)kc * FDIM);
      float4* dst = (float4*)lw;
#pragma unroll
      for (int t = 0; t < 8; ++t)
        dst[threadIdx.x + t * 256] = src[threadIdx.x + t * 256];
    }
#endif
    __syncthreads();

    // ---- 16 k-steps of 4 against the staged chunk
#pragma unroll 4
    for (int kb = 0; kb < KC; kb += 4) {
      const v2f a = *(const v2f*)(arow + kc + kb);        // A[r0+l][kc+kb+2*half+0/1]
      const float* wp = lw + (kb + 2 * half) * FDIM + l;
#pragma unroll
      for (int nt = 0; nt < 8; ++nt) {
        v2f b;
        b.x = wp[nt * 16];                                // W[kb+2*half  ][nt*16+l]
        b.y = wp[nt * 16 + FDIM];                         // W[kb+2*half+1][nt*16+l]
        acc[nt] = __builtin_amdgcn_wmma_f32_16x16x4_f32(
            false, a, false, b, (short)0, acc[nt], false, false);
      }
    }
  }

  if (valid) {
#pragma unroll
    for (int nt = 0; nt < 8; ++nt)
#pragma unroll
      for (int r = 0; r < 8; ++r)
        C[(r0 + r + 8 * half) * FDIM + nt * 16 + l] = acc[nt][r];
  }
}

// ---------------------------------------------------------------- edge aggregation
// agg[dst] += dinv[src]*w*dinv[dst] * xw[src];  one wave per edge, float4 per lane.
__global__ void __launch_bounds__(256)
aggregate_kernel(const float* __restrict__ xw, const long long* __restrict__ ei,
                 const float* __restrict__ ew, const float* __restrict__ dinv,
                 float* __restrict__ agg, int E) {
  int e = blockIdx.x * 8 + (threadIdx.x >> 5);
  if (e >= E) return;                       // wave-uniform
  int lane = threadIdx.x & 31;
  size_t s = (size_t)ei[e];
  size_t d = (size_t)ei[(size_t)E + e];
  float coef = dinv[s] * ew[e] * dinv[d];
  float4 v = ((const float4*)(xw + s * FDIM))[lane];
  float* dp = agg + d * FDIM + lane * 4;
  unsafeAtomicAdd(dp + 0, coef * v.x);
  unsafeAtomicAdd(dp + 1, coef * v.y);
  unsafeAtomicAdd(dp + 2, coef * v.z);
  unsafeAtomicAdd(dp + 3, coef * v.w);
}

// ---------------------------------------------------------------- GCN finish
// out = act( agg + dinv^2 * xw + b );  act: 0 = relu, 1 = sigmoid
__global__ void finish_kernel(const float* __restrict__ agg,
                              const float* __restrict__ xw,
                              const float* __restrict__ dinv,
                              const float* __restrict__ b,
                              float* __restrict__ out, long total, int act) {
  long i = (long)blockIdx.x * blockDim.x + threadIdx.x;
  if (i >= total) return;
  long row = i >> 7;
  int f = (int)(i & 127);
  float di = dinv[row];
  float v = agg[i] + di * di * xw[i] + b[f];
  out[i] = act ? (1.0f / (1.0f + expf(-v))) : fmaxf(v, 0.0f);
}

// ---------------------------------------------------------------- GRU glue
__global__ void concat_kernel(const float* __restrict__ x, const float* __restrict__ g,
                              const float* __restrict__ h, float* __restrict__ cat,
                              long total) {
  long i = (long)blockIdx.x * blockDim.x + threadIdx.x;
  if (i >= total) return;
  long row = i >> 7;
  int f = (int)(i & 127);
  float* c = cat + row * 384;
  c[f] = x[i];
  c[128 + f] = g[i];
  c[256 + f] = h[i];
}

// u = sigmoid(upre+bu) (stored in place); cat[:,256:384] = sigmoid(rpre+br) * h
__global__ void gate_kernel(float* __restrict__ upre, const float* __restrict__ rpre,
                            const float* __restrict__ h, const float* __restrict__ bu,
                            const float* __restrict__ br, float* __restrict__ cat,
                            long total) {
  long i = (long)blockIdx.x * blockDim.x + threadIdx.x;
  if (i >= total) return;
  long row = i >> 7;
  int f = (int)(i & 127);
  float u = 1.0f / (1.0f + expf(-(upre[i] + bu[f])));
  upre[i] = u;
  float r = 1.0f / (1.0f + expf(-(rpre[i] + br[f])));
  cat[row * 384 + 256 + f] = r * h[i];
}

// out = u*h + (1-u)*tanh(cpre+bc)
__global__ void final_kernel(const float* __restrict__ u, const float* __restrict__ cpre,
                             const float* __restrict__ h, const float* __restrict__ bc,
                             float* __restrict__ out, long total) {
  long i = (long)blockIdx.x * blockDim.x + threadIdx.x;
  if (i >= total) return;
  int f = (int)(i & 127);
  float c = tanhf(cpre[i] + bc[f]);
  float uu = u[i];
  out[i] = uu * h[i] + (1.0f - uu) * c;
}

// ---------------------------------------------------------------- launch
extern "C" void kernel_launch(void* const* d_in, const int* in_sizes, int n_in,
                              void* d_out, int out_size, void* d_ws, size_t ws_size,
                              hipStream_t stream) {
  const float*     x  = (const float*)d_in[0];
  const long long* ei = (const long long*)d_in[1];   // int64 (2,E)
  const float*     ew = (const float*)d_in[2];
  const float*     h  = (const float*)d_in[3];
  const float*     W1 = (const float*)d_in[4];
  const float*     b1 = (const float*)d_in[5];
  const float*     W2 = (const float*)d_in[6];
  const float*     b2 = (const float*)d_in[7];
  const float*     Wu = (const float*)d_in[8];
  const float*     bu = (const float*)d_in[9];
  const float*     Wr = (const float*)d_in[10];
  const float*     br = (const float*)d_in[11];
  const float*     Wc = (const float*)d_in[12];
  const float*     bc = (const float*)d_in[13];

  const int N = in_sizes[0] / FDIM;      // 50000
  const int E = in_sizes[2];             // 1,600,000
  const long NF = (long)N * FDIM;

  // workspace carve-out (~154 MB)
  float* ws   = (float*)d_ws;
  float* deg  = ws;                  // N
  float* dinv = deg + N;             // N
  float* t0   = dinv + N;            // N*128  (xw / u_pre / u)
  float* t1   = t0 + NF;             // N*128  (agg / r_pre / c_pre)
  float* g    = t1 + NF;             // N*128  (gcn output)
  float* cat  = g + NF;              // N*384

  const int TB = 256;
  dim3 blk(TB);
  unsigned gN   = (unsigned)((N + TB - 1) / TB);
  unsigned gE   = (unsigned)((E + TB - 1) / TB);
  unsigned gNF  = (unsigned)((NF + TB - 1) / TB);
  unsigned gZ   = (unsigned)((NF / 4 + TB - 1) / TB);
  unsigned gAgg = (unsigned)((E + 7) / 8);
  unsigned gGemm = (unsigned)((N / 16 + 7) / 8);   // 8 tiles (waves) per block

  // normalization
  init_deg_kernel<<<gN, blk, 0, stream>>>(deg, N);
  scatter_deg_kernel<<<gE, blk, 0, stream>>>(ei, ew, deg, E);
  rsqrt_kernel<<<gN, blk, 0, stream>>>(deg, dinv, N);

  // ---- GCN layer 1: g = relu(gcn(x, W1, b1))
  gemm_wmma_f32_kernel<<<gGemm, blk, 0, stream>>>(x, W1, t0, N, FDIM);
  zero4_kernel<<<gZ, blk, 0, stream>>>((float4*)t1, NF / 4);
  aggregate_kernel<<<gAgg, blk, 0, stream>>>(t0, ei, ew, dinv, t1, E);
  finish_kernel<<<gNF, blk, 0, stream>>>(t1, t0, dinv, b1, g, NF, 0);

  // ---- GCN layer 2: g = sigmoid(gcn(g, W2, b2))
  gemm_wmma_f32_kernel<<<gGemm, blk, 0, stream>>>(g, W2, t0, N, FDIM);
  zero4_kernel<<<gZ, blk, 0, stream>>>((float4*)t1, NF / 4);
  aggregate_kernel<<<gAgg, blk, 0, stream>>>(t0, ei, ew, dinv, t1, E);
  finish_kernel<<<gNF, blk, 0, stream>>>(t1, t0, dinv, b2, g, NF, 1);

  // ---- GRU gates
  concat_kernel<<<gNF, blk, 0, stream>>>(x, g, h, cat, NF);
  gemm_wmma_f32_kernel<<<gGemm, blk, 0, stream>>>(cat, Wu, t0, N, 3 * FDIM);
  gemm_wmma_f32_kernel<<<gGemm, blk, 0, stream>>>(cat, Wr, t1, N, 3 * FDIM);
  gate_kernel<<<gNF, blk, 0, stream>>>(t0, t1, h, bu, br, cat, NF);
  gemm_wmma_f32_kernel<<<gGemm, blk, 0, stream>>>(cat, Wc, t1, N, 3 * FDIM);
  final_kernel<<<gNF, blk, 0, stream>>>(t0, t1, h, bc, (float*)d_out, NF);
}